// EvolutionFeature_90529320665514
// MI455X (gfx1250) — compile-verified
//
#include <hip/hip_runtime.h>
#include <hip/hip_bf16.h>
#include <math.h>

#define NB 16
#define NC 256
#define HH 128
#define WWID 128
#define HWN (HH * WWID)
#define LCAP 512
#define NL 16
#define ISZ 512

typedef __attribute__((ext_vector_type(2))) float v2f;
typedef __attribute__((ext_vector_type(8))) float v8f;

// ---------------------------------------------------------------------------
// Wave-level 16x16 fp32 GEMM tile: C[m][n] = sum_k A[m][k] * Bt[n][k]
// A row-major [16, K] with leading dim lda; Bt row-major [16, K] (i.e. B^T).
// Uses V_WMMA_F32_16X16X4_F32 (wave32). EXEC must be all ones (uniform flow).
// A-layout: lane m = lane&15; VGPR0/1 hold k = kh, kh+1 with kh = 2*(lane>=16)
// B-layout: lane n = lane&15; same k pairing. C/D: vgpr r -> M = r + 8*(lane>=16).
// ---------------------------------------------------------------------------
__device__ __forceinline__ v8f wave_tile_gemm(const float* __restrict__ A, int lda,
                                              const float* __restrict__ Bt, int ldb,
                                              int K) {
  const int lane = threadIdx.x & 31;
  const int mn = lane & 15;
  const int kh = (lane >> 4) << 1;  // 0 or 2
  const float* ar = A + (size_t)mn * lda + kh;
  const float* br = Bt + (size_t)mn * ldb + kh;
  v8f acc;
#pragma unroll
  for (int i = 0; i < 8; ++i) acc[i] = 0.0f;
  for (int k0 = 0; k0 < K; k0 += 4) {
    v2f a = *(const v2f*)(ar + k0);
    v2f b = *(const v2f*)(br + k0);
    acc = __builtin_amdgcn_wmma_f32_16x16x4_f32(false, a, false, b, (short)0, acc,
                                                false, false);
  }
  return acc;
}

// ---------------------------------------------------------------------------
// K0: zero the histogram accumulator
// ---------------------------------------------------------------------------
__global__ void zero_hist_kernel(float* hist) {
  int i = threadIdx.x;
  if (i < NB * 2 * NL) hist[i] = 0.0f;
}

// ---------------------------------------------------------------------------
// K1: foreground mask (P1 > P0) + ordered stream compaction (first LCAP idxs)
// one block (8 waves) per batch; also writes ~valid output.
// ---------------------------------------------------------------------------
__global__ void compact_kernel(const float* __restrict__ P, int* __restrict__ idx,
                               int* __restrict__ nvalid, float* __restrict__ out_nv) {
  int b = blockIdx.x;
  const float* P0 = P + (size_t)b * 2 * HWN;
  const float* P1 = P0 + HWN;
  __shared__ int wcnt[8];
  __shared__ int s_base;
  if (threadIdx.x == 0) s_base = 0;
  __syncthreads();
  int wave = threadIdx.x >> 5;
  int lane = threadIdx.x & 31;
  for (int t0 = 0; t0 < HWN; t0 += 256) {
    int t = t0 + threadIdx.x;
    bool pred = P1[t] > P0[t];
    unsigned bal = __builtin_amdgcn_ballot_w32(pred);
    int lpre = __builtin_popcount(bal & ((1u << lane) - 1u));
    if (lane == 0) wcnt[wave] = __builtin_popcount(bal);
    __syncthreads();
    int wpre = 0;
    for (int w = 0; w < wave; ++w) wpre += wcnt[w];
    int pos = s_base + wpre + lpre;
    if (pred && pos < LCAP) idx[b * LCAP + pos] = t;
    __syncthreads();
    if (threadIdx.x == 0) {
      int tot = 0;
      for (int w = 0; w < 8; ++w) tot += wcnt[w];
      s_base += tot;
    }
    __syncthreads();
  }
  int n = s_base < LCAP ? s_base : LCAP;
  if (threadIdx.x == 0) nvalid[b] = n;
  for (int p = threadIdx.x; p < LCAP; p += 256) {
    out_nv[b * LCAP + p] = (p < n) ? 0.0f : 1.0f;  // ~valid
    if (p >= n) idx[b * LCAP + p] = 0;
  }
}

// ---------------------------------------------------------------------------
// K2: per (batch, token-slot) gather:
//   TO  = f_q rows (zeroed when invalid), tnO = row-normalized TO
//   TI  = antialiased 4x triangle resample of I_q at token pixel; tnI normalized
//   FI  = TI @ W_I^T + b_I  (zeroed when invalid) -> first half of X
// ---------------------------------------------------------------------------
__global__ void gather_kernel(const float* __restrict__ f_q, const float* __restrict__ I_q,
                              const int* __restrict__ idx, const int* __restrict__ nvalid,
                              const float* __restrict__ W_I, const float* __restrict__ b_I,
                              float* __restrict__ TO, float* __restrict__ tnO,
                              float* __restrict__ tnI, float* __restrict__ X) {
  int b = blockIdx.y, pos = blockIdx.x;
  int n = nvalid[b];
  bool valid = pos < n;
  int t = valid ? idx[b * LCAP + pos] : 0;
  int c = threadIdx.x;

  float v = valid ? f_q[((size_t)b * NC + c) * HWN + t] : 0.0f;

  __shared__ float red[256];
  red[c] = v * v;
  __syncthreads();
  for (int s = 128; s > 0; s >>= 1) {
    if (c < s) red[c] += red[c + s];
    __syncthreads();
  }
  float inv = 1.0f / fmaxf(sqrtf(red[0]), 1e-12f);
  size_t row = ((size_t)b * LCAP + pos) * NC;
  TO[row + c] = v;
  tnO[row + c] = v * inv;

  // jax.image.resize(bilinear, antialias) 512->128: 8-tap triangle, sep.
  __shared__ float ti[4];
  const float wt[8] = {0.03125f, 0.09375f, 0.15625f, 0.21875f,
                       0.21875f, 0.15625f, 0.09375f, 0.03125f};
  if (c < 3) {
    float acc = 0.0f;
    if (valid) {
      int h = t >> 7, w = t & 127;
      const float* img = I_q + ((size_t)b * 3 + c) * ISZ * ISZ;
      for (int dy = 0; dy < 8; ++dy) {
        int y = 4 * h - 2 + dy;
        y = y < 0 ? 0 : (y > ISZ - 1 ? ISZ - 1 : y);
        float rowacc = 0.0f;
        for (int dx = 0; dx < 8; ++dx) {
          int x = 4 * w - 2 + dx;
          x = x < 0 ? 0 : (x > ISZ - 1 ? ISZ - 1 : x);
          rowacc += wt[dx] * img[(size_t)y * ISZ + x];
        }
        acc += wt[dy] * rowacc;
      }
    }
    ti[c] = acc;
  }
  __syncthreads();
  if (c < 3) {
    float nr = sqrtf(ti[0] * ti[0] + ti[1] * ti[1] + ti[2] * ti[2]);
    tnI[((size_t)b * LCAP + pos) * 3 + c] = ti[c] / fmaxf(nr, 1e-12f);
  }
  float fi = valid ? (ti[0] * W_I[c * 3 + 0] + ti[1] * W_I[c * 3 + 1] +
                      ti[2] * W_I[c * 3 + 2] + b_I[c])
                   : 0.0f;
  X[((size_t)b * LCAP + pos) * (2 * NC) + c] = fi;
}

// ---------------------------------------------------------------------------
// K3: FO_tok = TO @ W_O^T + b_O (rows >= n zeroed) -> second half of X
// 4 wave-tiles per block; 512 tiles (32 M x 16 N) per batch.
// ---------------------------------------------------------------------------
__global__ void fo_gemm_kernel(const float* __restrict__ TO, const float* __restrict__ W_O,
                               const float* __restrict__ b_O, const int* __restrict__ nvalid,
                               float* __restrict__ X) {
  int b = blockIdx.y;
  int tile = blockIdx.x * 4 + (threadIdx.x >> 5);
  int tm = tile >> 4, tn = tile & 15;
  const float* A = TO + ((size_t)b * LCAP + tm * 16) * NC;
  const float* Bt = W_O + (size_t)tn * 16 * NC;
  v8f acc = wave_tile_gemm(A, NC, Bt, NC, NC);
  int lane = threadIdx.x & 31;
  int nn = lane & 15, mb = (lane >> 4) << 3;
  int n = nvalid[b];
  float bias = b_O[tn * 16 + nn];
#pragma unroll
  for (int r = 0; r < 8; ++r) {
    int m = tm * 16 + mb + r;
    float val = (m < n) ? acc[r] + bias : 0.0f;
    X[((size_t)b * LCAP + m) * (2 * NC) + NC + tn * 16 + nn] = val;
  }
}

// ---------------------------------------------------------------------------
// K4: Gram GO = tnO @ tnO^T (WMMA), A = clip((g+1)/2,0,1), triangular soft
// histogram over valid pairs -> hist[b][16..31] (LDS ds_add_f32 then global).
// ---------------------------------------------------------------------------
__global__ void gram_hist_kernel(const float* __restrict__ tnO,
                                 const int* __restrict__ nvalid,
                                 float* __restrict__ hist) {
  int b = blockIdx.y;
  int tile = blockIdx.x * 4 + (threadIdx.x >> 5);
  int tm = tile >> 5, tn = tile & 31;
  __shared__ float h[NL];
  if (threadIdx.x < NL) h[threadIdx.x] = 0.0f;
  __syncthreads();
  const float* A = tnO + ((size_t)b * LCAP + tm * 16) * NC;
  const float* Bt = tnO + ((size_t)b * LCAP + tn * 16) * NC;
  v8f acc = wave_tile_gemm(A, NC, Bt, NC, NC);
  int n = nvalid[b];
  int lane = threadIdx.x & 31;
  int nn = lane & 15, mb = (lane >> 4) << 3;
  int j = tn * 16 + nn;
  bool jv = j < n;
#pragma unroll
  for (int r = 0; r < 8; ++r) {
    int i = tm * 16 + mb + r;
    if (jv && i < n) {
      float Aff = fminf(fmaxf((acc[r] + 1.0f) * 0.5f, 0.0f), 1.0f);
      float f = Aff * (float)NL - 0.5f;
      int j0 = (int)floorf(f);
      float w1 = f - (float)j0;
      float w0 = 1.0f - w1;
      if (j0 >= 0 && j0 < NL) atomicAdd(&h[j0], w0);
      if (j0 + 1 >= 0 && j0 + 1 < NL) atomicAdd(&h[j0 + 1], w1);
    }
  }
  __syncthreads();
  if (threadIdx.x < NL) atomicAdd(&hist[b * 2 * NL + NL + threadIdx.x], h[threadIdx.x]);
}

// ---------------------------------------------------------------------------
// K5: GI histogram (K=3 dots, VALU) -> hist[b][0..15]
// ---------------------------------------------------------------------------
__global__ void gi_hist_kernel(const float* __restrict__ tnI,
                               const int* __restrict__ nvalid, float* __restrict__ hist) {
  int b = blockIdx.x;
  int n = nvalid[b];
  __shared__ float h[NL];
  if (threadIdx.x < NL) h[threadIdx.x] = 0.0f;
  __syncthreads();
  const float* T = tnI + (size_t)b * LCAP * 3;
  int total = n * n;
  for (int p = threadIdx.x; p < total; p += 256) {
    int i = p / n, j = p - i * n;
    float d = T[i * 3] * T[j * 3] + T[i * 3 + 1] * T[j * 3 + 1] +
              T[i * 3 + 2] * T[j * 3 + 2];
    float Aff = fminf(fmaxf((d + 1.0f) * 0.5f, 0.0f), 1.0f);
    float f = Aff * (float)NL - 0.5f;
    int j0 = (int)floorf(f);
    float w1 = f - (float)j0;
    float w0 = 1.0f - w1;
    if (j0 >= 0 && j0 < NL) atomicAdd(&h[j0], w0);
    if (j0 + 1 >= 0 && j0 + 1 < NL) atomicAdd(&h[j0 + 1], w1);
  }
  __syncthreads();
  if (threadIdx.x < NL) atomicAdd(&hist[b * 2 * NL + threadIdx.x], h[threadIdx.x]);
}

// ---------------------------------------------------------------------------
// K6: per-batch Es MLP: normalize 2 hist halves, Linear(32,256)+ReLU, Linear(256,256)
// ---------------------------------------------------------------------------
__global__ void es_mlp_kernel(const float* __restrict__ hist, const float* __restrict__ W1,
                              const float* __restrict__ b1, const float* __restrict__ W2,
                              const float* __restrict__ b2, float* __restrict__ E_s) {
  int b = blockIdx.x;
  __shared__ float h[2 * NL];
  __shared__ float hid[NC];
  if (threadIdx.x < 2 * NL) h[threadIdx.x] = hist[b * 2 * NL + threadIdx.x];
  __syncthreads();
  float s0 = 0.0f, s1 = 0.0f;
  for (int i = 0; i < NL; ++i) { s0 += h[i]; s1 += h[NL + i]; }
  s0 = fmaxf(s0, 1.0f);
  s1 = fmaxf(s1, 1.0f);
  int c = threadIdx.x;
  float acc = b1[c];
  for (int k = 0; k < NL; ++k) acc += (h[k] / s0) * W1[c * 2 * NL + k];
  for (int k = NL; k < 2 * NL; ++k) acc += (h[k] / s1) * W1[c * 2 * NL + k];
  hid[c] = fmaxf(acc, 0.0f);
  __syncthreads();
  float acc2 = b2[c];
  for (int k = 0; k < NC; ++k) acc2 += hid[k] * W2[c * NC + k];
  E_s[b * NC + c] = acc2;
}

// ---------------------------------------------------------------------------
// K7: Ep layer1: H1 = relu(X @ Ep_W1^T + b1), M=512 K=512 N=256 (WMMA)
// ---------------------------------------------------------------------------
__global__ void ep1_gemm_kernel(const float* __restrict__ X, const float* __restrict__ W1,
                                const float* __restrict__ b1, float* __restrict__ H1) {
  int b = blockIdx.y;
  int tile = blockIdx.x * 4 + (threadIdx.x >> 5);
  int tm = tile >> 4, tn = tile & 15;
  const float* A = X + ((size_t)b * LCAP + tm * 16) * (2 * NC);
  const float* Bt = W1 + (size_t)tn * 16 * (2 * NC);
  v8f acc = wave_tile_gemm(A, 2 * NC, Bt, 2 * NC, 2 * NC);
  int lane = threadIdx.x & 31;
  int nn = lane & 15, mb = (lane >> 4) << 3;
  float bias = b1[tn * 16 + nn];
#pragma unroll
  for (int r = 0; r < 8; ++r)
    H1[((size_t)b * LCAP + tm * 16 + mb + r) * NC + tn * 16 + nn] =
        fmaxf(acc[r] + bias, 0.0f);
}

// ---------------------------------------------------------------------------
// K8: Ep layer2 + E_s broadcast: E = H1 @ Ep_W2^T + b2 + E_s[b] -> d_out (WMMA)
// ---------------------------------------------------------------------------
__global__ void ep2_gemm_kernel(const float* __restrict__ H1, const float* __restrict__ W2,
                                const float* __restrict__ b2, const float* __restrict__ E_s,
                                float* __restrict__ out) {
  int b = blockIdx.y;
  int tile = blockIdx.x * 4 + (threadIdx.x >> 5);
  int tm = tile >> 4, tn = tile & 15;
  const float* A = H1 + ((size_t)b * LCAP + tm * 16) * NC;
  const float* Bt = W2 + (size_t)tn * 16 * NC;
  v8f acc = wave_tile_gemm(A, NC, Bt, NC, NC);
  int lane = threadIdx.x & 31;
  int nn = lane & 15, mb = (lane >> 4) << 3;
  int col = tn * 16 + nn;
  float add = b2[col] + E_s[b * NC + col];
#pragma unroll
  for (int r = 0; r < 8; ++r)
    out[((size_t)b * LCAP + tm * 16 + mb + r) * NC + col] = acc[r] + add;
}

// ---------------------------------------------------------------------------
extern "C" void kernel_launch(void* const* d_in, const int* in_sizes, int n_in,
                              void* d_out, int out_size, void* d_ws, size_t ws_size,
                              hipStream_t stream) {
  const float* I_q   = (const float*)d_in[0];
  const float* f_q   = (const float*)d_in[1];
  const float* P     = (const float*)d_in[2];
  const float* W_I   = (const float*)d_in[3];
  const float* b_I   = (const float*)d_in[4];
  const float* W_O   = (const float*)d_in[5];
  const float* b_O   = (const float*)d_in[6];
  const float* Ep_W1 = (const float*)d_in[7];
  const float* Ep_b1 = (const float*)d_in[8];
  const float* Ep_W2 = (const float*)d_in[9];
  const float* Ep_b2 = (const float*)d_in[10];
  const float* Es_W1 = (const float*)d_in[11];
  const float* Es_b1 = (const float*)d_in[12];
  const float* Es_W2 = (const float*)d_in[13];
  const float* Es_b2 = (const float*)d_in[14];

  float* out = (float*)d_out;
  float* out_nv = out + (size_t)NB * LCAP * NC;  // ~valid, flat after E

  // workspace layout (floats; all offsets even -> 8B aligned for b64 loads)
  float* w = (float*)d_ws;
  int* idx      = (int*)w;                               // NB*LCAP ints
  int* nvalid   = (int*)(w + NB * LCAP);                 // NB ints (pad 16)
  float* hist   = w + NB * LCAP + 16;                    // NB*32
  float* E_s    = hist + NB * 2 * NL;                    // NB*NC
  float* TO     = E_s + NB * NC;                         // NB*LCAP*NC (reused as H1)
  float* tnO    = TO + (size_t)NB * LCAP * NC;           // NB*LCAP*NC
  float* tnI    = tnO + (size_t)NB * LCAP * NC;          // NB*LCAP*3 (even count)
  float* X      = tnI + (size_t)NB * LCAP * 4;           // NB*LCAP*2C (pad tnI to 4/row)
  float* H1     = TO;                                    // reuse: TO consumed before K7

  zero_hist_kernel<<<1, 512, 0, stream>>>(hist);
  compact_kernel<<<NB, 256, 0, stream>>>(P, idx, nvalid, out_nv);
  gather_kernel<<<dim3(LCAP, NB), 256, 0, stream>>>(f_q, I_q, idx, nvalid, W_I, b_I,
                                                    TO, tnO, tnI, X);
  fo_gemm_kernel<<<dim3(128, NB), 128, 0, stream>>>(TO, W_O, b_O, nvalid, X);
  gram_hist_kernel<<<dim3(256, NB), 128, 0, stream>>>(tnO, nvalid, hist);
  gi_hist_kernel<<<NB, 256, 0, stream>>>(tnI, nvalid, hist);
  es_mlp_kernel<<<NB, 256, 0, stream>>>(hist, Es_W1, Es_b1, Es_W2, Es_b2, E_s);
  ep1_gemm_kernel<<<dim3(128, NB), 128, 0, stream>>>(X, Ep_W1, Ep_b1, H1);
  ep2_gemm_kernel<<<dim3(128, NB), 128, 0, stream>>>(H1, Ep_W2, Ep_b2, E_s, out);
}